// Decoder_45286135169607
// MI455X (gfx1250) — compile-verified
//
#include <hip/hip_runtime.h>

typedef __attribute__((ext_vector_type(2))) float v2f;
typedef __attribute__((ext_vector_type(8))) float v8f;

namespace {
constexpr int kEdges = 600000;
constexpr int kD    = 128;   // input feature dim
constexpr int kH1   = 64;    // hidden 1
constexpr int kH2   = 32;    // hidden 2
constexpr int kTileM = 16;   // edges per wave-tile (WMMA M)
constexpr int kWaves = 16;   // waves per block (512 threads) -> VGPR cap = file/4
constexpr int kTiles = kEdges / kTileM;   // 37500, exact
constexpr int kLdsA  = 132;  // padded A row stride (floats) -> conflict-free ds_load_b64
constexpr int kLdsH1 = 68;   // padded h1 row stride
constexpr int kLdsH2 = 36;   // padded h2 row stride
constexpr int kH1Base = kTileM * kLdsA;            // 2112: h1 region (no A aliasing:
constexpr int kH2Base = kH1Base + kTileM * kLdsH1; // 3200   async echo reads A region
constexpr int kScr    = kH2Base + kTileM * kLdsH2; // 3776   during compute)
constexpr int kGrid  = 1172; // ~2 tiles per wave; one ~283KB workgroup per WGP
}

__global__ __launch_bounds__(kWaves * 32)
void decoder_mlp_wmma(const float* __restrict__ X,       // [100000,128] node features
                      const int*   __restrict__ srcIdx,  // [E]
                      const float* __restrict__ labels,  // [E]
                      const float* __restrict__ W1,      // [128,64] row-major
                      const float* __restrict__ b1,      // [64]
                      const float* __restrict__ W2,      // [64,32]
                      const float* __restrict__ b2,      // [32]
                      const float* __restrict__ W3,      // [32]
                      const float* __restrict__ b3,      // [1]
                      float* __restrict__ out)           // logits[E] ++ labels[E] ++ emb[E*128]
{
  // Weights stored PRE-TRANSPOSED into per-lane WMMA B-fragment order:
  // sW1t[((k*4+n)*32 + lane)*2 + j] = W1[(4k + (lane>>4)*2 + j)][n*16 + (lane&15)]
  // -> each B fragment is ONE conflict-free ds_load_b64 per lane, no packing movs.
  __shared__ float sW1t[kD * kH1];
  __shared__ float sW2t[kH1 * kH2];
  __shared__ float sW3[kH2];
  __shared__ float sb1[kH1];
  __shared__ float sb2[kH2];
  __shared__ float sb3[1];
  __shared__ float sScr[kWaves][kScr];   // 16*15104B + weights ~= 283KB < 320KB/WGP

  const int t = threadIdx.x;
  for (int i = t; i < kD * kH1; i += kWaves * 32) {
    const int j  = i & 1;
    const int ln = (i >> 1) & 31;
    const int n  = (i >> 6) & 3;
    const int k  = i >> 8;                       // 0..31
    const int row = 4 * k + ((ln >> 4) << 1) + j;
    const int col = n * 16 + (ln & 15);
    sW1t[i] = W1[row * kH1 + col];
  }
  for (int i = t; i < kH1 * kH2; i += kWaves * 32) {
    const int j  = i & 1;
    const int ln = (i >> 1) & 31;
    const int n  = (i >> 6) & 1;
    const int k  = i >> 7;                       // 0..15
    const int row = 4 * k + ((ln >> 4) << 1) + j;
    const int col = n * 16 + (ln & 15);
    sW2t[i] = W2[row * kH2 + col];
  }
  if (t < kH1) sb1[t] = b1[t];
  if (t < kH2) { sb2[t] = b2[t]; sW3[t] = W3[t]; }
  if (t == 0) sb3[0] = b3[0];
  __syncthreads();

  const int wave = t >> 5;
  const int lane = t & 31;
  float* scr = sScr[wave];
  // LDS byte address of this lane's 16B slot in row 0 of the per-wave A region.
  // Generic LDS pointers carry the workgroup-relative LDS offset in addr[31:0].
  const unsigned scrAddr = (unsigned)(uintptr_t)scr + (unsigned)lane * 16u;

  const int mrow  = lane & 15;          // M row (A) / N col (B,C,D) within 16
  const int khalf = (lane >> 4) << 1;   // ISA A 16x4 layout: lanes16-31 hold K+2/3
  const int mhi   = (lane >> 4) << 3;   // ISA C/D layout: lanes16-31 hold M+8
  const float* w1f = sW1t + lane * 2;   // lane-fixed fragment base
  const float* w2f = sW2t + lane * 2;

  float* outLogits = out;
  float* outLabels = out + kEdges;
  float* outEmb    = out + 2L * kEdges;

  const v8f vzero = {0.f, 0.f, 0.f, 0.f, 0.f, 0.f, 0.f, 0.f};

  for (int tile = blockIdx.x * kWaves + wave; tile < kTiles;
       tile += gridDim.x * kWaves) {
    const long e0 = (long)tile * kTileM;

    int myIdx = 0;
    if (lane < 16) myIdx = srcIdx[e0 + lane];

    // Previous tile's async echo stores must finish reading the A region
    // before we overwrite it (async LDS accesses are unordered). They had the
    // whole previous compute phase to drain, so this is normally free.
    asm volatile("s_wait_asynccnt 0x0" ::: "memory");

    // ---- gather: 16 async global->LDS row copies, ALL in flight (ASYNCcnt),
    //      zero data VGPRs; one coalesced 512B row per instruction ----
#pragma unroll
    for (int r = 0; r < kTileM; ++r) {
      const int node = __builtin_amdgcn_readlane(myIdx, r);
      const float* gsrc = X + (long)node * kD + lane * 4;
      const unsigned dsa = scrAddr + (unsigned)(r * (kLdsA * 4));
      asm volatile("global_load_async_to_lds_b128 %0, %1, off"
                   :: "v"(dsa), "v"(gsrc) : "memory");
    }
    asm volatile("s_wait_asynccnt 0x0" ::: "memory");

    // ---- node_emb echo: async LDS->global stores; they drain in the
    //      background while the WMMA phase below executes ----
#pragma unroll
    for (int r = 0; r < kTileM; ++r) {
      float* gdst = outEmb + (e0 + r) * kD + lane * 4;
      const unsigned dsa = scrAddr + (unsigned)(r * (kLdsA * 4));
      asm volatile("global_store_async_from_lds_b128 %0, %1, off"
                   :: "v"(gdst), "v"(dsa) : "memory");
    }

    // ---- layer 1: [16x128] x [128x64] via v_wmma_f32_16x16x4_f32 ----
    v8f c[4];
#pragma unroll
    for (int n = 0; n < 4; ++n) c[n] = vzero;
    for (int k = 0; k < kD / 4; ++k) {
      const v2f a = *(const v2f*)(scr + mrow * kLdsA + 4 * k + khalf);
#pragma unroll
      for (int n = 0; n < 4; ++n) {
        const v2f b = *(const v2f*)(w1f + (k * 4 + n) * 64);
        c[n] = __builtin_amdgcn_wmma_f32_16x16x4_f32(
            false, a, false, b, (short)0, c[n], false, false);
      }
    }
    // bias + relu, spill h1 [16x64] to its own LDS region
#pragma unroll
    for (int n = 0; n < 4; ++n) {
      const float bias = sb1[n * 16 + mrow];
#pragma unroll
      for (int v = 0; v < 8; ++v) {
        scr[kH1Base + (v + mhi) * kLdsH1 + n * 16 + mrow] =
            fmaxf(c[n][v] + bias, 0.f);
      }
    }
    asm volatile("s_wait_dscnt 0" ::: "memory");

    // ---- layer 2: [16x64] x [64x32] ----
    v8f d2[2];
    d2[0] = vzero; d2[1] = vzero;
    for (int k = 0; k < kH1 / 4; ++k) {
      const v2f a = *(const v2f*)(scr + kH1Base + mrow * kLdsH1 + 4 * k + khalf);
#pragma unroll
      for (int n = 0; n < 2; ++n) {
        const v2f b = *(const v2f*)(w2f + (k * 2 + n) * 64);
        d2[n] = __builtin_amdgcn_wmma_f32_16x16x4_f32(
            false, a, false, b, (short)0, d2[n], false, false);
      }
    }
#pragma unroll
    for (int n = 0; n < 2; ++n) {
      const float bias = sb2[n * 16 + mrow];
#pragma unroll
      for (int v = 0; v < 8; ++v) {
        scr[kH2Base + (v + mhi) * kLdsH2 + n * 16 + mrow] =
            fmaxf(d2[n][v] + bias, 0.f);
      }
    }
    asm volatile("s_wait_dscnt 0" ::: "memory");

    // ---- layer 3: per-edge 32-dot + outputs (logits, labels copy) ----
    if (lane < 16) {
      const float* h2 = scr + kH2Base + lane * kLdsH2;
      float acc = sb3[0];
#pragma unroll
      for (int j = 0; j < kH2; ++j) acc = fmaf(h2[j], sW3[j], acc);
      const long e = e0 + lane;
      outLogits[e] = acc;
      outLabels[e] = labels[e];
    }
  }
}

extern "C" void kernel_launch(void* const* d_in, const int* in_sizes, int n_in,
                              void* d_out, int out_size, void* d_ws, size_t ws_size,
                              hipStream_t stream) {
  (void)in_sizes; (void)n_in; (void)out_size; (void)d_ws; (void)ws_size;
  const float* X      = (const float*)d_in[0];
  const int*   srcIdx = (const int*)  d_in[1];
  const float* labels = (const float*)d_in[2];
  const float* W1     = (const float*)d_in[3];
  const float* b1     = (const float*)d_in[4];
  const float* W2     = (const float*)d_in[5];
  const float* b2     = (const float*)d_in[6];
  const float* W3     = (const float*)d_in[7];
  const float* b3     = (const float*)d_in[8];
  float* out = (float*)d_out;

  decoder_mlp_wmma<<<dim3(kGrid), dim3(kWaves * 32), 0, stream>>>(
      X, srcIdx, labels, W1, b1, W2, b2, W3, b3, out);
}